// DWTLoss_32083405701424
// MI455X (gfx1250) — compile-verified
//
#include <hip/hip_runtime.h>

typedef float v2f __attribute__((ext_vector_type(2)));
typedef float v4f __attribute__((ext_vector_type(4)));
typedef float v8f __attribute__((ext_vector_type(8)));

#define GRID_MAX 2048
#define BLOCK    256   // 8 wave32s on CDNA5

// ---------------------------------------------------------------------------
// Pass 1: streaming Haar-DWT abs-sum. Each task = one float4 of a top row +
// the float4 directly below it = two 2x2 Haar blocks (64B of input per task).
// Accumulate |LL'|+|LH'|+|HL'|+|HH'| (pre-0.5-scale) of (pred-target).
// ---------------------------------------------------------------------------
__global__ __launch_bounds__(BLOCK) void dwt_loss_partial(
    const float* __restrict__ pred, const float* __restrict__ targ,
    float* __restrict__ partials,
    unsigned ntask,     // rowPairs * quadsPerRow
    unsigned qshift,    // log2(quadsPerRow)
    unsigned qmask,     // quadsPerRow - 1
    unsigned W)         // row length in floats
{
    const unsigned stride = gridDim.x * blockDim.x;
    float acc = 0.0f;

    for (unsigned t = blockIdx.x * blockDim.x + threadIdx.x; t < ntask; t += stride) {
        const unsigned q = t & qmask;        // quad-column within row pair
        const unsigned r = t >> qshift;      // row-pair index (global, row-major)
        const unsigned top = r * 2u * W + q * 4u;

        // Streamed once: non-temporal b128 loads (201MB > 192MB L2, no reuse).
        v4f p0 = __builtin_nontemporal_load((const v4f*)(pred + top));
        v4f p1 = __builtin_nontemporal_load((const v4f*)(pred + top + W));
        v4f g0 = __builtin_nontemporal_load((const v4f*)(targ + top));
        v4f g1 = __builtin_nontemporal_load((const v4f*)(targ + top + W));

        v4f e0 = p0 - g0;   // a0 b0 a1 b1   (top of two 2x2 blocks)
        v4f e1 = p1 - g1;   // c0 d0 c1 d1   (bottom)

        // Block 0
        float t0 = e0.x + e0.y, t1 = e1.x + e1.y;
        float t2 = e0.x - e0.y, t3 = e1.x - e1.y;
        acc += fabsf(t0 + t1) + fabsf(t0 - t1) + fabsf(t2 + t3) + fabsf(t2 - t3);
        // Block 1
        t0 = e0.z + e0.w; t1 = e1.z + e1.w;
        t2 = e0.z - e0.w; t3 = e1.z - e1.w;
        acc += fabsf(t0 + t1) + fabsf(t0 - t1) + fabsf(t2 + t3) + fabsf(t2 - t3);
    }

    // ---- Wave32 reduction via one V_WMMA_F32_16X16X4_F32 ----
    // A(16x4): lanes 0-15 supply K=0, lanes 16-31 supply K=2 (VGPR0 = a.x);
    // a.y (K=1/K=3) zeroed. B(4x16) = ones. D[m][n] = rowsum(m).
    // Per-lane sum of D's 8 VGPRs: lanes 0-15 hold sum(rows 0-7),
    // lanes 16-31 hold sum(rows 8-15); one shfl_xor(16) completes the wave sum.
    v2f a; a.x = acc;  a.y = 0.0f;
    v2f b; b.x = 1.0f; b.y = 1.0f;
    v8f c = {0.f, 0.f, 0.f, 0.f, 0.f, 0.f, 0.f, 0.f};
    v8f d = __builtin_amdgcn_wmma_f32_16x16x4_f32(
        /*neg_a=*/false, a, /*neg_b=*/false, b,
        /*c_mod=*/(short)0, c, /*reuse_a=*/false, /*reuse_b=*/false);
    float ws = d[0] + d[1] + d[2] + d[3] + d[4] + d[5] + d[6] + d[7];
    ws += __shfl_xor(ws, 16, 32);

    // ---- Cross-wave (deterministic, fixed order) ----
    __shared__ float wpart[BLOCK / 32];
    const unsigned lane = threadIdx.x & 31u;
    const unsigned wid  = threadIdx.x >> 5u;
    if (lane == 0) wpart[wid] = ws;
    __syncthreads();
    if (threadIdx.x == 0) {
        float bs = 0.0f;
        for (int i = 0; i < BLOCK / 32; ++i) bs += wpart[i];
        partials[blockIdx.x] = bs;
    }
}

// ---------------------------------------------------------------------------
// Pass 2: deterministic final reduction of per-block partials + scale.
// ---------------------------------------------------------------------------
__global__ __launch_bounds__(256) void dwt_loss_finalize(
    const float* __restrict__ partials, int nparts,
    float* __restrict__ out, float scale)
{
    __shared__ float sm[256];
    float s = 0.0f;
    for (int i = threadIdx.x; i < nparts; i += 256) s += partials[i];
    sm[threadIdx.x] = s;
    __syncthreads();
    for (int off = 128; off > 0; off >>= 1) {
        if ((int)threadIdx.x < off) sm[threadIdx.x] += sm[threadIdx.x + off];
        __syncthreads();
    }
    if (threadIdx.x == 0) out[0] = sm[0] * scale;
}

extern "C" void kernel_launch(void* const* d_in, const int* in_sizes, int n_in,
                              void* d_out, int out_size, void* d_ws, size_t ws_size,
                              hipStream_t stream) {
    const float* pred = (const float*)d_in[0];
    const float* targ = (const float*)d_in[1];
    float* out = (float*)d_out;
    float* partials = (float*)d_ws;

    const unsigned n = (unsigned)in_sizes[0];      // B*C*H*W = 25,165,824
    const unsigned W = 512u;                       // reference inner width
    const unsigned rows = n / W;                   // B*C*H (rows are contiguous)
    const unsigned rowPairs = rows / 2u;           // H even per image -> pairs never straddle
    const unsigned quads = W / 4u;                 // float4s per row
    unsigned qshift = 0; while ((1u << qshift) < quads) ++qshift;
    const unsigned qmask = quads - 1u;
    const unsigned ntask = rowPairs * quads;

    int grid = GRID_MAX;
    if (ws_size < (size_t)grid * sizeof(float))    // defensive clamp to scratch
        grid = (int)(ws_size / sizeof(float));
    if (grid < 1) grid = 1;

    dwt_loss_partial<<<grid, BLOCK, 0, stream>>>(pred, targ, partials,
                                                 ntask, qshift, qmask, W);
    // loss = 2/n * sum of pre-scale abs coefficients (0.5 coeff scale folded in)
    dwt_loss_finalize<<<1, 256, 0, stream>>>(partials, grid, out, 2.0f / (float)n);
}